// PyrmCNet_52682068853286
// MI455X (gfx1250) — compile-verified
//
#include <hip/hip_runtime.h>
#include <math.h>

// ---------------- problem constants (from reference) ----------------
#define BGRAPH 16
#define NP0 25000
#define NP1 20000
#define NP2 12000
#define KSEL0 20000   // ceil(0.8*25000)
#define KSEL1 12000   // ceil(0.6*20000)
#define KSEL2 4800    // ceil(0.4*12000)
#define NEDGE 3200000
#define NTOT0 (BGRAPH*NP0)    // 400000
#define NTOT1 (BGRAPH*KSEL0)  // 320000
#define NTOT2 (BGRAPH*KSEL1)  // 192000
#define NTOT3 (BGRAPH*KSEL2)  // 76800
#define BN_EPS 1e-5f

typedef __attribute__((ext_vector_type(2))) float v2f;
typedef __attribute__((ext_vector_type(8))) float v8f;

// ---------------- CDNA5 f32 WMMA (16x16x4) ----------------
// A: 16x4 f32, lanes 0-15 hold {K=0,K=1} for M=lane, lanes 16-31 hold {K=2,K=3}
// B: 4x16 f32, mirrored layout (N=lane&15)
// C/D: 8 VGPRs; vgpr r -> (M=r, N=lane) lanes 0-15 ; (M=8+r, N=lane-16) lanes 16-31
static __device__ __forceinline__ v8f wmma4(v2f a, v2f b, v8f c) {
#if __has_builtin(__builtin_amdgcn_wmma_f32_16x16x4_f32)
  return __builtin_amdgcn_wmma_f32_16x16x4_f32(false, a, false, b, (short)0, c,
                                               false, false);
#else
  // host-pass / non-gfx1250 placeholder only
  c[0] += a[0] * b[0];
  return c;
#endif
}

static __device__ __forceinline__ void atomicAddF(float* p, float v) {
  (void)__hip_atomic_fetch_add(p, v, __ATOMIC_RELAXED, __HIP_MEMORY_SCOPE_AGENT);
}

// ---------------- utility kernels ----------------
__global__ void fill_f32(float* p, float v, int n) {
  int i = blockIdx.x * blockDim.x + threadIdx.x;
  if (i < n) p[i] = v;
}
__global__ void fill_i32(int* p, int v, int n) {
  int i = blockIdx.x * blockDim.x + threadIdx.x;
  if (i < n) p[i] = v;
}
__global__ void copy_edges(const int* __restrict__ ei, int* __restrict__ s,
                           int* __restrict__ d, int* __restrict__ m) {
  int e = blockIdx.x * blockDim.x + threadIdx.x;
  if (e >= NEDGE) return;
  s[e] = ei[e];
  d[e] = ei[NEDGE + e];
  m[e] = 1;
}

// ---------------- stage-0 MLP (cin = 5) ----------------
__global__ void mlp5_kernel(const float* __restrict__ x, const float* __restrict__ w,
                            const float* __restrict__ b, float* __restrict__ y, int n) {
  __shared__ float sw[5 * 64];
  __shared__ float sb[64];
  for (int i = threadIdx.x; i < 5 * 64; i += blockDim.x) sw[i] = w[i];
  if (threadIdx.x < 64) sb[threadIdx.x] = b[threadIdx.x];
  __syncthreads();
  int idx = blockIdx.x * blockDim.x + threadIdx.x;
  if (idx >= n * 64) return;
  int row = idx >> 6, c = idx & 63;
  const float* xr = x + (size_t)row * 5;
  float acc = sb[c];
#pragma unroll
  for (int k = 0; k < 5; ++k) acc = fmaf(xr[k], sw[k * 64 + c], acc);
  y[idx] = acc;
}

// ---------------- f32-WMMA GEMM: Y[n,64] = X[n,64] @ W[64,64] (+bias) ----------------
// 256 threads = 8 waves; wave handles a 16-row tile x 64 cols (4 x v8f accum).
// REQUIREMENT: n % 128 == 0 (all call sites: 400000 / 320000 / 192000) -> no guards,
// straight-line WMMA loop (EXEC stays all-ones as WMMA requires) and branch-free stores.
__global__ void gemm_n64k64(const float* __restrict__ X, const float* __restrict__ W,
                            const float* __restrict__ bias, float* __restrict__ Y,
                            int n) {
  (void)n;
  __shared__ float sW[64 * 64];
  __shared__ float sB[64];
  for (int i = threadIdx.x; i < 4096; i += blockDim.x) sW[i] = W[i];
  if (threadIdx.x < 64) sB[threadIdx.x] = bias ? bias[threadIdx.x] : 0.0f;
  __syncthreads();
  const int wave = threadIdx.x >> 5, lane = threadIdx.x & 31;
  const int row0 = blockIdx.x * 128 + wave * 16;
  const int mrow = row0 + (lane & 15);
  const int kh = (lane >> 4) * 2;  // lanes 16-31 carry K+2,K+3
  const int col = lane & 15;
  const float* xr = X + (size_t)mrow * 64;
  v8f acc[4] = {};
  for (int kt = 0; kt < 16; ++kt) {
    v2f a;
    a.x = xr[kt * 4 + kh];
    a.y = xr[kt * 4 + kh + 1];
#pragma unroll
    for (int nt = 0; nt < 4; ++nt) {
      v2f bf;
      bf.x = sW[(kt * 4 + kh) * 64 + nt * 16 + col];
      bf.y = sW[(kt * 4 + kh + 1) * 64 + nt * 16 + col];
      acc[nt] = wmma4(a, bf, acc[nt]);
    }
  }
  const int rbase = row0 + ((lane < 16) ? 0 : 8);
#pragma unroll
  for (int nt = 0; nt < 4; ++nt) {
    const float bb = sB[nt * 16 + col];
#pragma unroll
    for (int r = 0; r < 8; ++r)
      Y[(size_t)(rbase + r) * 64 + nt * 16 + col] = acc[nt][r] + bb;
  }
}

// ---------------- BatchNorm (training-mode, biased var) ----------------
__global__ void bn_stats(const float* __restrict__ x, int n, float* __restrict__ acc) {
  int c = threadIdx.x & 63;
  int slice = blockIdx.x * (blockDim.x >> 6) + (threadIdx.x >> 6);
  int nslice = gridDim.x * (blockDim.x >> 6);
  float s = 0.f, q = 0.f;
  for (int r = slice; r < n; r += nslice) {
    float v = x[(size_t)r * 64 + c];
    s += v;
    q = fmaf(v, v, q);
  }
  atomicAddF(&acc[c], s);
  atomicAddF(&acc[64 + c], q);
}
__global__ void bn_finalize(const float* __restrict__ acc, float n,
                            const float* __restrict__ g, const float* __restrict__ b,
                            float* __restrict__ ab) {
  int c = threadIdx.x;
  if (c >= 64) return;
  float mean = acc[c] / n;
  float var = acc[64 + c] / n - mean * mean;
  float s = g[c] * rsqrtf(var + BN_EPS);
  ab[c] = s;
  ab[64 + c] = b[c] - mean * s;
}
__global__ void bn_apply(float* __restrict__ x, int total, const float* __restrict__ ab,
                         int dorelu) {
  int idx = blockIdx.x * blockDim.x + threadIdx.x;
  if (idx >= total) return;
  int c = idx & 63;
  float v = fmaf(x[idx], ab[c], ab[64 + c]);
  x[idx] = dorelu ? fmaxf(v, 0.0f) : v;
}

// ---------------- GCN pieces ----------------
__global__ void edge_deg(const int* __restrict__ dst, const int* __restrict__ em,
                         float* __restrict__ deg) {
  int e = blockIdx.x * blockDim.x + threadIdx.x;
  if (e >= NEDGE || !em[e]) return;
  atomicAddF(&deg[dst[e]], 1.0f);
}
__global__ void rsqrt_kernel(const float* __restrict__ deg, float* __restrict__ dinv,
                             int n) {
  int i = blockIdx.x * blockDim.x + threadIdx.x;
  if (i < n) dinv[i] = rsqrtf(deg[i]);
}
// one edge handled by 16 threads (float4 each): agg[dst] += xw[src] * dinv[s]*dinv[d]
__global__ void edge_agg(const int* __restrict__ src, const int* __restrict__ dst,
                         const int* __restrict__ em, const float* __restrict__ xw,
                         const float* __restrict__ dinv, float* __restrict__ agg) {
  size_t t = (size_t)blockIdx.x * blockDim.x + threadIdx.x;
  if (t >= (size_t)NEDGE * 16) return;
  int e = (int)(t >> 4), q = (int)(t & 15);
  if (!em[e]) return;
  int s = src[e], d = dst[e];
  float coef = dinv[s] * dinv[d];
  const float4 v = ((const float4*)(xw + (size_t)s * 64))[q];
  float* ap = agg + (size_t)d * 64 + q * 4;
  atomicAddF(ap + 0, v.x * coef);
  atomicAddF(ap + 1, v.y * coef);
  atomicAddF(ap + 2, v.z * coef);
  atomicAddF(ap + 3, v.w * coef);
}
// out = agg + xw/deg + bias   (self-loop term)
__global__ void gcn_combine(float* __restrict__ a, const float* __restrict__ xw,
                            const float* __restrict__ deg, const float* __restrict__ bias,
                            int n) {
  int idx = blockIdx.x * blockDim.x + threadIdx.x;
  if (idx >= n * 64) return;
  int row = idx >> 6, c = idx & 63;
  a[idx] = a[idx] + xw[idx] / deg[row] + bias[c];
}

// ---------------- TopK pooling ----------------
__global__ void wnorm_kernel(const float* __restrict__ wp, float* __restrict__ wn) {
  __shared__ float red[64];
  float v = wp[threadIdx.x];
  red[threadIdx.x] = v * v;
  __syncthreads();
  for (int off = 32; off > 0; off >>= 1) {
    if (threadIdx.x < off) red[threadIdx.x] += red[threadIdx.x + off];
    __syncthreads();
  }
  if (threadIdx.x == 0) wn[0] = sqrtf(red[0]);
}
__global__ void score_kernel(const float* __restrict__ h, const float* __restrict__ wp,
                             const float* __restrict__ wn, float* __restrict__ score,
                             int n) {
  __shared__ float sw[64];
  __shared__ float inorm;
  if (threadIdx.x < 64) sw[threadIdx.x] = wp[threadIdx.x];
  if (threadIdx.x == 0) inorm = 1.0f / wn[0];
  __syncthreads();
  int i = blockIdx.x * blockDim.x + threadIdx.x;
  if (i >= n) return;
  const float* hr = h + (size_t)i * 64;
  float d = 0.f;
#pragma unroll
  for (int c = 0; c < 64; ++c) d = fmaf(hr[c], sw[c], d);
  score[i] = tanhf(d * inorm);
}
// per-graph threshold so |{score > thr}| == k  (float bisection; set-selection is
// order-invariant for this net since the head is a mean pool)
__global__ void topk_thresh(const float* __restrict__ score, int np, int k,
                            float* __restrict__ thr) {
  __shared__ int red[256];
  __shared__ float slo, shi;
  int g = blockIdx.x;
  const float* s = score + (size_t)g * np;
  if (threadIdx.x == 0) { slo = -1.0001f; shi = 1.0001f; }
  __syncthreads();
  for (int it = 0; it < 48; ++it) {
    float mid = 0.5f * (slo + shi);
    int c = 0;
    for (int i = threadIdx.x; i < np; i += 256) c += (s[i] > mid) ? 1 : 0;
    red[threadIdx.x] = c;
    __syncthreads();
    for (int off = 128; off > 0; off >>= 1) {
      if (threadIdx.x < off) red[threadIdx.x] += red[threadIdx.x + off];
      __syncthreads();
    }
    if (threadIdx.x == 0) {
      if (red[0] >= k) slo = mid; else shi = mid;
    }
    __syncthreads();
  }
  if (threadIdx.x == 0) thr[g] = slo;
}
__global__ void topk_select(const float* __restrict__ score, const float* __restrict__ thr,
                            int np, int k, int ntot, int* __restrict__ newid,
                            int* __restrict__ perm, int* __restrict__ cnt) {
  int i = blockIdx.x * blockDim.x + threadIdx.x;
  if (i >= ntot) return;
  int g = i / np;
  int id = -1;
  if (score[i] > thr[g]) {
    int pos = atomicAdd(&cnt[g], 1);
    if (pos < k) {
      id = g * k + pos;
      perm[id] = i;
    }
  }
  newid[i] = id;
}
__global__ void gather_gate(const float* __restrict__ h, const float* __restrict__ score,
                            const int* __restrict__ perm, int nnew,
                            float* __restrict__ out) {
  int t = blockIdx.x * blockDim.x + threadIdx.x;
  if (t >= nnew * 16) return;
  int j = t >> 4, q = t & 15;
  int i = perm[j];
  float s = score[i];
  float4 v = ((const float4*)(h + (size_t)i * 64))[q];
  float4 r;
  r.x = v.x * s; r.y = v.y * s; r.z = v.z * s; r.w = v.w * s;
  ((float4*)(out + (size_t)j * 64))[q] = r;
}
__global__ void relabel_edges(int* __restrict__ src, int* __restrict__ dst,
                              int* __restrict__ em, const int* __restrict__ newid) {
  int e = blockIdx.x * blockDim.x + threadIdx.x;
  if (e >= NEDGE) return;
  int ns = newid[src[e]], nd = newid[dst[e]];
  int m = em[e] & (ns >= 0 ? 1 : 0) & (nd >= 0 ? 1 : 0);
  src[e] = ns < 0 ? 0 : ns;
  dst[e] = nd < 0 ? 0 : nd;
  em[e] = m;
}

// ---------------- global mean pool over the 3 pooled stages ----------------
__global__ void meanpool_kernel(const float* __restrict__ xs0,
                                const float* __restrict__ xs1,
                                const float* __restrict__ xs2, float* __restrict__ xg) {
  __shared__ float red[256];
  int g = blockIdx.x, c = threadIdx.x & 63, p = threadIdx.x >> 6;
  float s = 0.f;
  for (int r = p; r < KSEL0; r += 4) s += xs0[((size_t)g * KSEL0 + r) * 64 + c];
  for (int r = p; r < KSEL1; r += 4) s += xs1[((size_t)g * KSEL1 + r) * 64 + c];
  for (int r = p; r < KSEL2; r += 4) s += xs2[((size_t)g * KSEL2 + r) * 64 + c];
  red[threadIdx.x] = s;
  __syncthreads();
  if (p == 0)
    xg[g * 64 + c] =
        (red[c] + red[64 + c] + red[128 + c] + red[192 + c]) * (1.0f / 36800.0f);
}

// ---------------- decoder layer 1 via WMMA: P[16,512] = xg[16,64]@w1 + b1 ----------------
__global__ void dec1_wmma(const float* __restrict__ xg, const float* __restrict__ w1,
                          const float* __restrict__ b1, float* __restrict__ Pm) {
  __shared__ float sX[16 * 64];
  for (int i = threadIdx.x; i < 1024; i += 256) sX[i] = xg[i];
  __syncthreads();
  int wave = threadIdx.x >> 5, lane = threadIdx.x & 31;
  int mr = lane & 15, kh = (lane >> 4) * 2, col = lane & 15;
  v8f acc[4] = {};
  for (int kt = 0; kt < 16; ++kt) {
    v2f a;
    a.x = sX[mr * 64 + kt * 4 + kh];
    a.y = sX[mr * 64 + kt * 4 + kh + 1];
#pragma unroll
    for (int t = 0; t < 4; ++t) {
      int nt = wave * 4 + t;
      v2f bf;
      bf.x = w1[(size_t)(kt * 4 + kh) * 512 + nt * 16 + col];
      bf.y = w1[(size_t)(kt * 4 + kh + 1) * 512 + nt * 16 + col];
      acc[t] = wmma4(a, bf, acc[t]);
    }
  }
  const int rbase = (lane < 16) ? 0 : 8;
#pragma unroll
  for (int t = 0; t < 4; ++t) {
    int c = (wave * 4 + t) * 16 + col;
    const float bb = b1[c];
#pragma unroll
    for (int r = 0; r < 8; ++r) Pm[(rbase + r) * 512 + c] = acc[t][r] + bb;
  }
}
__global__ void dec_bn(const float* __restrict__ Pm, const float* __restrict__ g,
                       const float* __restrict__ b, float* __restrict__ H) {
  int c = threadIdx.x;  // 512 threads
  float s = 0.f, q = 0.f;
  for (int r = 0; r < 16; ++r) {
    float v = Pm[r * 512 + c];
    s += v;
    q = fmaf(v, v, q);
  }
  float mean = s * (1.0f / 16.0f);
  float var = q * (1.0f / 16.0f) - mean * mean;
  float a = g[c] * rsqrtf(var + BN_EPS);
  float sh = b[c] - mean * a;
  for (int r = 0; r < 16; ++r) {
    float v = fmaf(Pm[r * 512 + c], a, sh);
    H[r * 512 + c] = fmaxf(v, 0.0f);
  }
}
__global__ void dec2_kernel(const float* __restrict__ H, const float* __restrict__ w0,
                            const float* __restrict__ b0, float* __restrict__ out) {
  __shared__ float sh[512];
  int g = blockIdx.x;
  for (int i = threadIdx.x; i < 512; i += blockDim.x) sh[i] = H[g * 512 + i];
  __syncthreads();
  for (int c = threadIdx.x; c < 3750; c += blockDim.x) {
    float acc = b0[c];
    for (int k = 0; k < 512; ++k) acc = fmaf(sh[k], w0[(size_t)k * 3750 + c], acc);
    out[(size_t)g * 3750 + c] = acc;
  }
}

// ---------------- host orchestration ----------------
extern "C" void kernel_launch(void* const* d_in, const int* in_sizes, int n_in,
                              void* d_out, int out_size, void* d_ws, size_t ws_size,
                              hipStream_t stream) {
  (void)in_sizes; (void)n_in; (void)out_size; (void)ws_size;
  const float* x = (const float*)d_in[0];
  const int* ei = (const int*)d_in[1];
  auto P = [&](int i) { return (const float*)d_in[i]; };

  // workspace bump allocator (256B aligned)
  char* wp = (char*)d_ws;
  auto alloc = [&](size_t bytes) -> void* {
    void* p = (void*)wp;
    wp += (bytes + 255) & ~(size_t)255;
    return p;
  };
  float* xs0  = (float*)alloc((size_t)NTOT1 * 64 * 4);
  float* xs1  = (float*)alloc((size_t)NTOT2 * 64 * 4);
  float* xs2  = (float*)alloc((size_t)NTOT3 * 64 * 4);
  float* A    = (float*)alloc((size_t)NTOT0 * 64 * 4);
  float* XW   = (float*)alloc((size_t)NTOT0 * 64 * 4);
  float* deg  = (float*)alloc((size_t)NTOT0 * 4);
  float* dinv = (float*)alloc((size_t)NTOT0 * 4);
  float* score= (float*)alloc((size_t)NTOT0 * 4);
  int* newid  = (int*)alloc((size_t)NTOT0 * 4);
  int* perm   = (int*)alloc((size_t)NTOT1 * 4);
  int* srcw   = (int*)alloc((size_t)NEDGE * 4);
  int* dstw   = (int*)alloc((size_t)NEDGE * 4);
  int* emw    = (int*)alloc((size_t)NEDGE * 4);
  float* bnacc= (float*)alloc(128 * 4);
  float* bnab = (float*)alloc(128 * 4);
  float* thr  = (float*)alloc(16 * 4);
  int* cnt    = (int*)alloc(16 * 4);
  float* wn   = (float*)alloc(256);
  float* xg   = (float*)alloc(16 * 64 * 4);
  float* Pd   = (float*)alloc(16 * 512 * 4);
  float* Hd   = (float*)alloc(16 * 512 * 4);

  copy_edges<<<(NEDGE + 255) / 256, 256, 0, stream>>>(ei, srcw, dstw, emw);

  const int nps[3] = {NP0, NP1, NP2};
  const int ks[3] = {KSEL0, KSEL1, KSEL2};
  float* xsarr[3] = {xs0, xs1, xs2};
  const float* Xin = x;

  for (int s = 0; s < 3; ++s) {
    int pb = 3 + s * 9;
    int np = nps[s], k = ks[s], n = BGRAPH * np;
    int tot = n * 64;
    // 1) MLP
    if (s == 0)
      mlp5_kernel<<<(tot + 255) / 256, 256, 0, stream>>>(Xin, P(pb + 0), P(pb + 1), A, n);
    else
      gemm_n64k64<<<n / 128, 256, 0, stream>>>(Xin, P(pb + 0), P(pb + 1), A, n);
    // 2) BN1 + ReLU
    fill_f32<<<1, 128, 0, stream>>>(bnacc, 0.0f, 128);
    bn_stats<<<1024, 256, 0, stream>>>(A, n, bnacc);
    bn_finalize<<<1, 64, 0, stream>>>(bnacc, (float)n, P(pb + 2), P(pb + 3), bnab);
    bn_apply<<<(tot + 255) / 256, 256, 0, stream>>>(A, tot, bnab, 1);
    // 3) GCNConv
    gemm_n64k64<<<n / 128, 256, 0, stream>>>(A, P(pb + 4), nullptr, XW, n);
    fill_f32<<<(n + 255) / 256, 256, 0, stream>>>(deg, 1.0f, n);  // self-loop
    edge_deg<<<(NEDGE + 255) / 256, 256, 0, stream>>>(dstw, emw, deg);
    rsqrt_kernel<<<(n + 255) / 256, 256, 0, stream>>>(deg, dinv, n);
    fill_f32<<<(tot + 255) / 256, 256, 0, stream>>>(A, 0.0f, tot);
    {
      size_t nt = (size_t)NEDGE * 16;
      edge_agg<<<(unsigned)((nt + 255) / 256), 256, 0, stream>>>(srcw, dstw, emw, XW,
                                                                 dinv, A);
    }
    gcn_combine<<<(tot + 255) / 256, 256, 0, stream>>>(A, XW, deg, P(pb + 5), n);
    // 4) BN2 + ReLU
    fill_f32<<<1, 128, 0, stream>>>(bnacc, 0.0f, 128);
    bn_stats<<<1024, 256, 0, stream>>>(A, n, bnacc);
    bn_finalize<<<1, 64, 0, stream>>>(bnacc, (float)n, P(pb + 6), P(pb + 7), bnab);
    bn_apply<<<(tot + 255) / 256, 256, 0, stream>>>(A, tot, bnab, 1);
    // 5) TopK pool
    wnorm_kernel<<<1, 64, 0, stream>>>(P(pb + 8), wn);
    score_kernel<<<(n + 255) / 256, 256, 0, stream>>>(A, P(pb + 8), wn, score, n);
    topk_thresh<<<BGRAPH, 256, 0, stream>>>(score, np, k, thr);
    fill_i32<<<1, 16, 0, stream>>>(cnt, 0, 16);
    topk_select<<<(n + 255) / 256, 256, 0, stream>>>(score, thr, np, k, n, newid, perm,
                                                     cnt);
    gather_gate<<<(BGRAPH * k * 16 + 255) / 256, 256, 0, stream>>>(A, score, perm,
                                                                   BGRAPH * k, xsarr[s]);
    relabel_edges<<<(NEDGE + 255) / 256, 256, 0, stream>>>(srcw, dstw, emw, newid);
    Xin = xsarr[s];
  }

  meanpool_kernel<<<BGRAPH, 256, 0, stream>>>(xs0, xs1, xs2, xg);
  dec1_wmma<<<1, 256, 0, stream>>>(xg, P(30), P(31), Pd);
  dec_bn<<<1, 512, 0, stream>>>(Pd, P(32), P(33), Hd);
  dec2_kernel<<<BGRAPH, 256, 0, stream>>>(Hd, P(34), P(35), (float*)d_out);
}